// GlobalPooling_37692632990102
// MI455X (gfx1250) — compile-verified
//
#include <hip/hip_runtime.h>

typedef __attribute__((ext_vector_type(2))) float v2f;
typedef __attribute__((ext_vector_type(8))) float v8f;

#define DCOLS 128
#define THREADS 256   // 8 waves of 32; wave w owns columns [16w, 16w+16)

__device__ __forceinline__ int lower_bound_i32(const int* __restrict__ b, int n, int key) {
    int lo = 0, hi = n;
    while (lo < hi) {
        int mid = (lo + hi) >> 1;
        if (b[mid] < key) lo = mid + 1; else hi = mid;
    }
    return lo;
}

__global__ __launch_bounds__(THREADS)
void GlobalPooling_seg_kernel(const float* __restrict__ x,
                              const int* __restrict__ batch,
                              float* __restrict__ out, int n) {
    const int seg   = blockIdx.x;
    // All threads compute both bounds redundantly (uniform addresses -> L2-resident, cheap).
    const int start = lower_bound_i32(batch, n, seg);
    const int end   = lower_bound_i32(batch, n, seg + 1);
    const int count = end - start;

    const int wave = threadIdx.x >> 5;
    const int lane = threadIdx.x & 31;
    const int half = lane >> 4;              // which 16-lane half of the wave
    const int col  = lane & 15;              // N index within the 16x16 tile
    const int c0   = (wave << 4) + col;      // absolute column 0..127

    const float NEG_INF = -__builtin_inff();

    v2f a; a.x = 1.0f; a.y = 1.0f;           // ones A (16x4): every D row = column sums of B
    v8f acc = {};                            // f32 accumulator, zero-initialized
    float m0 = NEG_INF, m1 = NEG_INF, m2 = NEG_INF, m3 = NEG_INF;

    // ---- Steady state: full 8-row groups, unconditional NT loads, no branches ----
    // Lane base: row (start+half), column c0. Slots: rows +0, +2, +4, +6 (x4B*128 = 512B/row).
    const float* __restrict__ p = x + (size_t)(start + half) * DCOLS + c0;
    const int fullGroups = count >> 3;
#pragma unroll 2
    for (int g = 0; g < fullGroups; ++g) {
        float va = __builtin_nontemporal_load(p + 0 * 2 * DCOLS);
        float vb = __builtin_nontemporal_load(p + 1 * 2 * DCOLS);
        float vc = __builtin_nontemporal_load(p + 2 * 2 * DCOLS);
        float vd = __builtin_nontemporal_load(p + 3 * 2 * DCOLS);
        p += 8 * DCOLS;
        m0 = fmaxf(m0, va); m1 = fmaxf(m1, vb);
        m2 = fmaxf(m2, vc); m3 = fmaxf(m3, vd);
        v2f b0; b0.x = va; b0.y = vb;
        v2f b1; b1.x = vc; b1.y = vd;
        acc = __builtin_amdgcn_wmma_f32_16x16x4_f32(false, a, false, b0, (short)0, acc, false, false);
        acc = __builtin_amdgcn_wmma_f32_16x16x4_f32(false, a, false, b1, (short)0, acc, false, false);
    }

    // ---- Tail: at most 7 remaining rows, predicated, runs once ----
    const int r0 = start + (fullGroups << 3);
    if (r0 < end) {
        const int ra = r0 + half;
        const int rb = r0 + 2 + half;
        const int rc = r0 + 4 + half;
        const int rd = r0 + 6 + half;
        float va = 0.0f, vb = 0.0f, vc = 0.0f, vd = 0.0f;
        if (ra < end) { va = __builtin_nontemporal_load(&x[(size_t)ra * DCOLS + c0]); m0 = fmaxf(m0, va); }
        if (rb < end) { vb = __builtin_nontemporal_load(&x[(size_t)rb * DCOLS + c0]); m1 = fmaxf(m1, vb); }
        if (rc < end) { vc = __builtin_nontemporal_load(&x[(size_t)rc * DCOLS + c0]); m2 = fmaxf(m2, vc); }
        if (rd < end) { vd = __builtin_nontemporal_load(&x[(size_t)rd * DCOLS + c0]); m3 = fmaxf(m3, vd); }
        v2f b0; b0.x = va; b0.y = vb;
        v2f b1; b1.x = vc; b1.y = vd;
        // EXEC fully reconverged here (divergence was local to the loads above).
        acc = __builtin_amdgcn_wmma_f32_16x16x4_f32(false, a, false, b0, (short)0, acc, false, false);
        acc = __builtin_amdgcn_wmma_f32_16x16x4_f32(false, a, false, b1, (short)0, acc, false, false);
    }

    // Every D row is identical -> acc[0] in each lane == segment sum of column (lane&15)
    float colsum = acc[0];
    float mymax  = fmaxf(fmaxf(m0, m1), fmaxf(m2, m3));
    // Other half of the wave covered the complementary row residues of the same column.
    mymax = fmaxf(mymax, __shfl_xor(mymax, 16, 32));

    if (half == 0) {
        const float inv = 1.0f / (float)(count < 1 ? 1 : count);
        const size_t base = (size_t)seg * (2 * DCOLS);
        out[base + c0]         = colsum * inv;  // mean pool, cols 0..127
        out[base + DCOLS + c0] = mymax;         // max pool,  cols 128..255
    }
}

extern "C" void kernel_launch(void* const* d_in, const int* in_sizes, int n_in,
                              void* d_out, int out_size, void* d_ws, size_t ws_size,
                              hipStream_t stream) {
    const float* x     = (const float*)d_in[0];
    const int*   batch = (const int*)d_in[1];
    float*       out   = (float*)d_out;
    const int n            = in_sizes[1];          // number of rows / index entries
    const int num_segments = out_size / (2 * DCOLS);

    GlobalPooling_seg_kernel<<<num_segments, THREADS, 0, stream>>>(x, batch, out, n);
}